// GCN_45535243272486
// MI455X (gfx1250) — compile-verified
//
#include <hip/hip_runtime.h>
#include <hip/hip_bf16.h>

#define DIM 128
#define NCLS 10
#define NGRAPH 256

typedef __attribute__((ext_vector_type(2))) float v2f;
typedef __attribute__((ext_vector_type(8))) float v8f;

// ---------------------------------------------------------------- degree
__global__ void deg_accum_kernel(const int* __restrict__ col,
                                 float* __restrict__ deg, int E) {
    int e = blockIdx.x * blockDim.x + threadIdx.x;
    if (e < E) atomicAdd(&deg[col[e]], 1.0f);
}

__global__ void dinv_kernel(float* __restrict__ deg, int N) {
    int i = blockIdx.x * blockDim.x + threadIdx.x;
    if (i < N) deg[i] = rsqrtf(deg[i] + 1.0f);   // +1 = self loop; deg>=1 always
}

// ---------------------------------------------------------------- f32 WMMA GEMM
// Y[N x 128] = act(X)[N x 128] @ W[128 x 128]
// act = identity (layer 1) or relu(x + xbias) fused on the A-operand load
// (layer 2), which removes a standalone 51MB bias+relu sweep.
// One wave per 16x16 output tile; 8 waves/block cover the 8 column tiles of a
// 16-row strip. K swept 0..127 in steps of 4 via V_WMMA_F32_16X16X4_F32.
// A (16x4 f32): lane L holds M = L&15; VGPR0 = K=2*(L>>4), VGPR1 = K+1.
// B (4x16 f32): lane L holds N = L&15; VGPR0 = K=2*(L>>4), VGPR1 = K+1.
// C/D (16x16 f32): VGPR r: lanes 0-15 -> M=r, lanes 16-31 -> M=r+8; N = L&15.
template <bool FUSE_RELU>
__global__ void gemm_wmma_f32(const float* __restrict__ X,
                              const float* __restrict__ W,
                              const float* __restrict__ xbias,
                              float* __restrict__ Y, int N) {
    const int wave = threadIdx.x >> 5;
    const int lane = threadIdx.x & 31;
    const int nl   = lane & 15;
    const int kh   = lane >> 4;          // 0 or 1 (K half)
    const int m0   = blockIdx.x << 4;    // row tile
    const int n0   = wave << 4;          // column tile
    if (m0 >= N) return;                 // wave-uniform (whole block exits)

    const float* xr = X + (size_t)(m0 + nl) * DIM;   // A row for this lane
    const float* wc = W + n0 + nl;                   // B column for this lane

    v8f acc = {};
#pragma unroll 8
    for (int k0 = 0; k0 < DIM; k0 += 4) {
        const int ka = k0 + (kh << 1);
        float a0 = xr[ka];
        float a1 = xr[ka + 1];
        if (FUSE_RELU) {
            a0 = fmaxf(a0 + xbias[ka], 0.f);
            a1 = fmaxf(a1 + xbias[ka + 1], 0.f);
        }
        v2f a, b;
        a.x = a0;
        a.y = a1;
        b.x = wc[(size_t)ka * DIM];
        b.y = wc[(size_t)(ka + 1) * DIM];
        acc = __builtin_amdgcn_wmma_f32_16x16x4_f32(
            /*neg_a=*/false, a, /*neg_b=*/false, b,
            /*c_mod=*/(short)0, acc, /*reuse_a=*/false, /*reuse_b=*/false);
    }

    float* yr = Y + (size_t)(m0 + (kh << 3)) * DIM + n0 + nl;
#pragma unroll
    for (int r = 0; r < 8; ++r) yr[(size_t)r * DIM] = acc[r];
}

// ---------------------------------------------------------------- self-loop init
// OUT[n] = XW[n] * dinv[n]^2  (plain store: doubles as zero-init of aggregate)
__global__ void self_init_kernel(const float* __restrict__ XW,
                                 const float* __restrict__ dinv,
                                 float* __restrict__ OUT, int N) {
    long long t = (long long)blockIdx.x * blockDim.x + threadIdx.x;
    int n = (int)(t >> 5);
    if (n >= N) return;                              // wave-uniform
    n = __builtin_amdgcn_readfirstlane(n);           // scalarize uniform index
    int lane = threadIdx.x & 31;
    float di  = dinv[n];
    float nrm = di * di;
    float4 v = ((const float4*)(XW + (size_t)n * DIM))[lane];
    float4 o;
    o.x = v.x * nrm; o.y = v.y * nrm; o.z = v.z * nrm; o.w = v.w * nrm;
    ((float4*)(OUT + (size_t)n * DIM))[lane] = o;
}

// ---------------------------------------------------------------- edge scatter
// One wave (32 lanes * float4) per edge: OUT[col] += XW[row] * dinv[row]*dinv[col]
// Edge index / norm loads are wave-uniform -> force to the scalar path so the
// vector memory pipe only carries the 512B gather + the 128 f32 atomics.
__global__ void edge_scatter_kernel(const float* __restrict__ XW,
                                    const int* __restrict__ row,
                                    const int* __restrict__ col,
                                    const float* __restrict__ dinv,
                                    float* __restrict__ OUT, int E) {
    long long t = (long long)blockIdx.x * blockDim.x + threadIdx.x;
    int e = (int)(t >> 5);
    if (e >= E) return;                              // wave-uniform
    e = __builtin_amdgcn_readfirstlane(e);
    int lane = threadIdx.x & 31;
    int r = row[e], c = col[e];
    float nrm = dinv[r] * dinv[c];
    float4 v = ((const float4*)(XW + (size_t)r * DIM))[lane];
    float* dst = OUT + (size_t)c * DIM + (lane << 2);
    atomicAdd(dst + 0, v.x * nrm);
    atomicAdd(dst + 1, v.y * nrm);
    atomicAdd(dst + 2, v.z * nrm);
    atomicAdd(dst + 3, v.w * nrm);
}

// ---------------------------------------------------------------- pooling
__global__ void pool_cnt_kernel(const int* __restrict__ batch,
                                float* __restrict__ cnt, int N) {
    int i = blockIdx.x * blockDim.x + threadIdx.x;
    if (i < N) atomicAdd(&cnt[batch[i]], 1.0f);
}

// pooled[g] += relu(H[n] + bias)   (bias+relu of layer 2 fused into the pool)
__global__ void pool_sum_kernel(const float* __restrict__ H,
                                const int* __restrict__ batch,
                                const float* __restrict__ bias,
                                float* __restrict__ pooled, int N) {
    long long t = (long long)blockIdx.x * blockDim.x + threadIdx.x;
    int n = (int)(t >> 5);
    if (n >= N) return;                              // wave-uniform
    n = __builtin_amdgcn_readfirstlane(n);
    int lane = threadIdx.x & 31;
    int g = batch[n];
    int d = lane << 2;
    float4 v = ((const float4*)(H + (size_t)n * DIM))[lane];
    v.x = fmaxf(v.x + bias[d + 0], 0.f);
    v.y = fmaxf(v.y + bias[d + 1], 0.f);
    v.z = fmaxf(v.z + bias[d + 2], 0.f);
    v.w = fmaxf(v.w + bias[d + 3], 0.f);
    float* dst = pooled + (size_t)g * DIM + d;
    atomicAdd(dst + 0, v.x);
    atomicAdd(dst + 1, v.y);
    atomicAdd(dst + 2, v.z);
    atomicAdd(dst + 3, v.w);
}

// ---------------------------------------------------------------- final FFN
// out[g,c] = (pooled[g]/max(cnt[g],1)) . Wf[:,c] + bf[c]
__global__ void final_ffn_kernel(const float* __restrict__ pooled,
                                 const float* __restrict__ cnt,
                                 const float* __restrict__ Wf,
                                 const float* __restrict__ bf,
                                 float* __restrict__ out) {
    int t = blockIdx.x * blockDim.x + threadIdx.x;
    if (t >= NGRAPH * NCLS) return;
    int g = t / NCLS, c = t % NCLS;
    float inv = 1.0f / fmaxf(cnt[g], 1.0f);
    float s = bf[c];
    const float* pg = pooled + (size_t)g * DIM;
#pragma unroll 8
    for (int k = 0; k < DIM; ++k) s += pg[k] * inv * Wf[(size_t)k * NCLS + c];
    out[t] = s;
}

// ---------------------------------------------------------------- launch
extern "C" void kernel_launch(void* const* d_in, const int* in_sizes, int n_in,
                              void* d_out, int out_size, void* d_ws, size_t ws_size,
                              hipStream_t stream) {
    const float* x    = (const float*)d_in[0];
    const int*   eidx = (const int*)  d_in[1];
    const int*   batch= (const int*)  d_in[2];
    const float* W1   = (const float*)d_in[3];
    const float* b1   = (const float*)d_in[4];
    const float* W2   = (const float*)d_in[5];
    const float* b2   = (const float*)d_in[6];
    const float* Wf   = (const float*)d_in[7];
    const float* bf   = (const float*)d_in[8];
    float* out = (float*)d_out;

    const int N = in_sizes[0] / DIM;   // 100000
    const int E = in_sizes[1] / 2;     // 3200000
    const int* row  = eidx;            // edge_index[0]
    const int* colv = eidx + E;        // edge_index[1]

    // workspace layout: A | B | dinv | pooled | cnt  (~103 MB)
    char* ws = (char*)d_ws;
    const size_t feat_bytes = (size_t)N * DIM * sizeof(float);
    float* A      = (float*)ws;
    float* B      = (float*)(ws + feat_bytes);
    float* dinv   = (float*)(ws + 2 * feat_bytes);
    float* pooled = (float*)(ws + 2 * feat_bytes + (size_t)N * sizeof(float));
    float* cnt    = pooled + (size_t)NGRAPH * DIM;

    // zero deg + pooled + cnt in one contiguous async memset
    hipMemsetAsync(dinv, 0,
                   (size_t)N * sizeof(float) +
                   ((size_t)NGRAPH * DIM + NGRAPH) * sizeof(float), stream);

    const int TB = 256;
    const int gemmBlocks     = (N + 15) / 16;
    const long long nodeWarp = (long long)N * 32;
    const int nodeWaveBlocks = (int)((nodeWarp + TB - 1) / TB);
    const long long edgeWarp = (long long)E * 32;
    const int edgeWaveBlocks = (int)((edgeWarp + TB - 1) / TB);

    // normalization
    deg_accum_kernel<<<(E + TB - 1) / TB, TB, 0, stream>>>(colv, dinv, E);
    dinv_kernel<<<(N + TB - 1) / TB, TB, 0, stream>>>(dinv, N);

    // layer 1: A = x@W1 ; B = selfloop(A) + scatter(A)
    gemm_wmma_f32<false><<<gemmBlocks, TB, 0, stream>>>(x, W1, nullptr, A, N);
    self_init_kernel<<<nodeWaveBlocks, TB, 0, stream>>>(A, dinv, B, N);
    edge_scatter_kernel<<<edgeWaveBlocks, TB, 0, stream>>>(A, row, colv, dinv, B, E);

    // layer 2: A = relu(B + b1)@W2 (bias+relu fused into GEMM A-loads)
    gemm_wmma_f32<true><<<gemmBlocks, TB, 0, stream>>>(B, W2, b1, A, N);
    self_init_kernel<<<nodeWaveBlocks, TB, 0, stream>>>(A, dinv, B, N);
    edge_scatter_kernel<<<edgeWaveBlocks, TB, 0, stream>>>(A, row, colv, dinv, B, E);

    // mean pool (bias+relu of layer 2 fused) + classifier head
    pool_cnt_kernel<<<(N + TB - 1) / TB, TB, 0, stream>>>(batch, cnt, N);
    pool_sum_kernel<<<nodeWaveBlocks, TB, 0, stream>>>(B, batch, b2, pooled, N);
    final_ffn_kernel<<<(NGRAPH * NCLS + TB - 1) / TB, TB, 0, stream>>>(
        pooled, cnt, Wf, bf, out);
}